// EdgeDecoder_32323923870320
// MI455X (gfx1250) — compile-verified
//
#include <hip/hip_runtime.h>
#include <hip/hip_bf16.h>

typedef float v2f __attribute__((ext_vector_type(2)));
typedef float v8f __attribute__((ext_vector_type(8)));

#define HH 128        // hidden dim
#define TWOH 256      // concat dim (K)

// D = A(16x4) * B(4x16) + C(16x16), fp32 WMMA, wave32.
// A lanes 0-15: VGPR0=K0,VGPR1=K1 (M=lane); lanes 16-31: K2,K3.
// B lanes 0-15: VGPR0=K0,VGPR1=K1 (N=lane); lanes 16-31: K2,K3.
// C/D: VGPR v: lanes0-15 -> M=v, lanes16-31 -> M=v+8, N=lane&15.

__global__ __launch_bounds__(256) void edge_decoder_wmma(
    const float* __restrict__ z_user,
    const float* __restrict__ z_recipe,
    const int*   __restrict__ eli,    // [2,E] int32
    const float* __restrict__ W1,     // [256,128] row-major (K,N)
    const float* __restrict__ b1,     // [128]
    const float* __restrict__ W2,     // [128]
    const float* __restrict__ b2,     // [1]
    float*       __restrict__ out,    // [E]
    int E)
{
    // Swizzled W1 in LDS: index ((kk*8+nn)*32 + lane)*2 + v  maps to
    // W1[4*kk + 2*(lane>>4) + v][16*nn + (lane&15)]
    __shared__ float w1s[TWOH * HH];   // 128 KB

    const int tid = threadIdx.x;
    for (int f = tid; f < TWOH * HH; f += 256) {
        int v    = f & 1;
        int lane = (f >> 1) & 31;
        int nn   = (f >> 6) & 7;
        int kk   = f >> 9;
        int k = (kk << 2) + ((lane >> 4) << 1) + v;
        int n = (nn << 4) + (lane & 15);
        w1s[f] = W1[k * HH + n];
    }
    __syncthreads();

    const int lane = tid & 31;
    const int m    = lane & 15;   // edge slot within tile
    const int kh   = lane >> 4;   // K-pair selector

    // Per-lane bias / W2 fragments: index n = 16*nn + m matches C/D layout.
    float b1v[8], w2v[8];
#pragma unroll
    for (int nn = 0; nn < 8; ++nn) {
        b1v[nn] = b1[(nn << 4) + m];
        w2v[nn] = W2[(nn << 4) + m];
    }
    const float bias2 = b2[0];

    const int waveId     = blockIdx.x * 8 + (tid >> 5);
    const int wavesTotal = gridDim.x * 8;
    const int nTiles = (E + 15) >> 4;
    const int nPairs = (nTiles + 1) >> 1;

    for (int p = waveId; p < nPairs; p += wavesTotal) {
        const int tile0 = p * 2;
        const int tile1 = tile0 + 1;
        const int e0 = tile0 * 16 + m;
        const int e1 = tile1 * 16 + m;
        const int e0c = (e0 < E) ? e0 : 0;
        const int e1c = (e1 < E) ? e1 : 0;
        const int row0 = eli[e0c];
        const int col0 = eli[E + e0c];
        const int row1 = eli[e1c];
        const int col1 = eli[E + e1c];

        const float* pu0 = z_user   + (size_t)row0 * HH + (kh << 1);
        const float* pr0 = z_recipe + (size_t)col0 * HH + (kh << 1);
        const float* pu1 = z_user   + (size_t)row1 * HH + (kh << 1);
        const float* pr1 = z_recipe + (size_t)col1 * HH + (kh << 1);

        v8f acc0[8], acc1[8];
#pragma unroll
        for (int nn = 0; nn < 8; ++nn) {
            acc0[nn] = v8f{0.f,0.f,0.f,0.f,0.f,0.f,0.f,0.f};
            acc1[nn] = v8f{0.f,0.f,0.f,0.f,0.f,0.f,0.f,0.f};
        }

        const float* wsp = w1s + (lane << 1);   // advances 512 floats / K-step

        // K = 0..127 : user half of the concat
#pragma unroll 8
        for (int kk = 0; kk < 32; ++kk) {
            v2f a0 = *(const v2f*)(pu0 + (kk << 2));
            v2f a1 = *(const v2f*)(pu1 + (kk << 2));
            v2f bf[8];
#pragma unroll
            for (int nn = 0; nn < 8; ++nn)
                bf[nn] = *(const v2f*)(wsp + (kk << 9) + (nn << 6));
#pragma unroll
            for (int nn = 0; nn < 8; ++nn) {
                acc0[nn] = __builtin_amdgcn_wmma_f32_16x16x4_f32(
                    false, a0, false, bf[nn], (short)0, acc0[nn], false, false);
                acc1[nn] = __builtin_amdgcn_wmma_f32_16x16x4_f32(
                    false, a1, false, bf[nn], (short)0, acc1[nn], false, false);
            }
        }
        // K = 128..255 : recipe half of the concat
        const float* wsp2 = wsp + (32 << 9);
#pragma unroll 8
        for (int kk = 0; kk < 32; ++kk) {
            v2f a0 = *(const v2f*)(pr0 + (kk << 2));
            v2f a1 = *(const v2f*)(pr1 + (kk << 2));
            v2f bf[8];
#pragma unroll
            for (int nn = 0; nn < 8; ++nn)
                bf[nn] = *(const v2f*)(wsp2 + (kk << 9) + (nn << 6));
#pragma unroll
            for (int nn = 0; nn < 8; ++nn) {
                acc0[nn] = __builtin_amdgcn_wmma_f32_16x16x4_f32(
                    false, a0, false, bf[nn], (short)0, acc0[nn], false, false);
                acc1[nn] = __builtin_amdgcn_wmma_f32_16x16x4_f32(
                    false, a1, false, bf[nn], (short)0, acc1[nn], false, false);
            }
        }

        // Epilogue: h = relu(acc + b1); partial = sum_n h * W2[n]
        float s0[8], s1[8];
#pragma unroll
        for (int v = 0; v < 8; ++v) { s0[v] = 0.f; s1[v] = 0.f; }
#pragma unroll
        for (int nn = 0; nn < 8; ++nn) {
#pragma unroll
            for (int v = 0; v < 8; ++v) {
                float h0 = fmaxf(acc0[nn][v] + b1v[nn], 0.f);
                float h1 = fmaxf(acc1[nn][v] + b1v[nn], 0.f);
                s0[v] = fmaf(h0, w2v[nn], s0[v]);
                s1[v] = fmaf(h1, w2v[nn], s1[v]);
            }
        }
        // Reduce over the 16 N-lanes of each half (masks <16 stay in-half).
#pragma unroll
        for (int v = 0; v < 8; ++v) {
            float t0 = s0[v], t1 = s1[v];
#pragma unroll
            for (int off = 1; off < 16; off <<= 1) {
                t0 += __shfl_xor(t0, off);
                t1 += __shfl_xor(t1, off);
            }
            if (m == 0) {
                int o0 = tile0 * 16 + (kh << 3) + v;
                int o1 = tile1 * 16 + (kh << 3) + v;
                if (o0 < E) out[o0] = t0 + bias2;
                if (o1 < E) out[o1] = t1 + bias2;
            }
        }
    }
}

extern "C" void kernel_launch(void* const* d_in, const int* in_sizes, int n_in,
                              void* d_out, int out_size, void* d_ws, size_t ws_size,
                              hipStream_t stream) {
    (void)n_in; (void)d_ws; (void)ws_size; (void)out_size;
    const float* z_user   = (const float*)d_in[0];
    const float* z_recipe = (const float*)d_in[1];
    const int*   eli      = (const int*)d_in[2];
    const float* W1       = (const float*)d_in[3];
    const float* b1       = (const float*)d_in[4];
    const float* W2       = (const float*)d_in[5];
    const float* b2       = (const float*)d_in[6];
    float*       out      = (float*)d_out;

    const int E = in_sizes[2] / 2;
    const int nTiles = (E + 15) / 16;
    const int nPairs = (nTiles + 1) / 2;
    int blocks = nPairs < 512 ? nPairs : 512;
    if (blocks < 1) blocks = 1;

    hipLaunchKernelGGL(edge_decoder_wmma, dim3(blocks), dim3(256), 0, stream,
                       z_user, z_recipe, eli, W1, b1, W2, b2, out, E);
}